// AFGRLEncoder_old_2662879724174
// MI455X (gfx1250) — compile-verified
//
#include <hip/hip_runtime.h>
#include <hip/hip_bf16.h>

typedef __attribute__((ext_vector_type(2))) float v2f;
typedef __attribute__((ext_vector_type(8))) float v8f;

#define IN_CH  128
#define HID_CH 512

// ---------------------------------------------------------------------------
// Phase 1: degree (including self loop), then dinv = rsqrt(deg) in place.
// ---------------------------------------------------------------------------
__global__ void deg_init_kernel(float* __restrict__ deg, int n) {
    int i = blockIdx.x * blockDim.x + threadIdx.x;
    if (i < n) deg[i] = 1.0f;  // self-loop contribution
}

__global__ void deg_accum_kernel(const long long* __restrict__ dst,
                                 float* __restrict__ deg, int nE) {
    int e = blockIdx.x * blockDim.x + threadIdx.x;
    if (e < nE) atomicAdd(&deg[(size_t)dst[e]], 1.0f);
}

__global__ void deg_rsqrt_kernel(float* __restrict__ deg, int n) {
    int i = blockIdx.x * blockDim.x + threadIdx.x;
    if (i < n) deg[i] = rsqrtf(deg[i]);  // deg >= 1 always (self loop)
}

// ---------------------------------------------------------------------------
// Phase 2: agg[n] = x[n] * dinv[n]^2  (self loop), then scatter-add edges.
// One thread handles 4 contiguous channels (float4); 32 threads cover a node.
// ---------------------------------------------------------------------------
__global__ void self_init_kernel(const float* __restrict__ x,
                                 const float* __restrict__ dinv,
                                 float* __restrict__ agg, int nNodes) {
    unsigned t = blockIdx.x * blockDim.x + threadIdx.x;
    unsigned node = t >> 5;
    if (node >= (unsigned)nNodes) return;
    unsigned c = (t & 31u) << 2;
    float di = dinv[node];
    float s = di * di;
    const float4 v = *(const float4*)(x + (size_t)node * IN_CH + c);
    float4 o; o.x = v.x * s; o.y = v.y * s; o.z = v.z * s; o.w = v.w * s;
    *(float4*)(agg + (size_t)node * IN_CH + c) = o;
}

__global__ void edge_scatter_kernel(const float* __restrict__ x,
                                    const long long* __restrict__ src,
                                    const long long* __restrict__ dst,
                                    const float* __restrict__ dinv,
                                    float* __restrict__ agg, int nE) {
    unsigned t = blockIdx.x * blockDim.x + threadIdx.x;
    unsigned e = t >> 5;             // 32 threads per edge
    if (e >= (unsigned)nE) return;
    unsigned c = (t & 31u) << 2;     // 4 channels per thread
    size_t s = (size_t)src[e];
    size_t d = (size_t)dst[e];
    float norm = dinv[s] * dinv[d];
    const float4 v = *(const float4*)(x + s * IN_CH + c);
    float* ap = agg + d * IN_CH + c;
    atomicAdd(ap + 0, v.x * norm);
    atomicAdd(ap + 1, v.y * norm);
    atomicAdd(ap + 2, v.z * norm);
    atomicAdd(ap + 3, v.w * norm);
}

// ---------------------------------------------------------------------------
// Phase 3: out = PReLU(agg @ W + b) via V_WMMA_F32_16X16X4_F32 (full fp32).
//
// Block = 256 threads = 8 waves; each block owns a 16-row M tile.
// Wave w computes columns [w*64, w*64+64) as 4 accumulators of 16x16.
// A tile (16x128 f32, 8KB) is staged in LDS with padded stride (132 floats)
// to avoid bank conflicts on the per-k ds_load_b64 fragment reads.
//
// VGPR layouts per CDNA5 ISA §7.12.2:
//   A 16x4 : lanes 0-15 -> M=lane,    K={kk,kk+1}; lanes 16-31 -> K={kk+2,kk+3}
//   B 4x16 : lanes 0-15 -> N=lane,    K={kk,kk+1}; lanes 16-31 -> K={kk+2,kk+3}
//   C 16x16: vgpr r, lane<16 -> (M=r, N=lane); lane>=16 -> (M=r+8, N=lane-16)
// Requires nNodes % 16 == 0 (100000 = 6250*16). EXEC stays all-ones.
// ---------------------------------------------------------------------------
#define LDS_STRIDE 132  // 128 + 4 floats of padding

__global__ void gemm_prelu_kernel(const float* __restrict__ agg,
                                  const float* __restrict__ W,
                                  const float* __restrict__ bias,
                                  const float* __restrict__ alpha_p,
                                  float* __restrict__ out, int nNodes) {
    __shared__ float As[16 * LDS_STRIDE];

    const int row0 = blockIdx.x * 16;
    const int tid  = threadIdx.x;

    // Stage the 16x128 A tile into LDS: 512 float4s, 2 per thread.
    for (int i = tid; i < 512; i += 256) {
        int r  = i >> 5;           // row 0..15
        int c4 = (i & 31) << 2;    // col 0..124 step 4
        const float4 v = *(const float4*)(agg + (size_t)(row0 + r) * IN_CH + c4);
        *(float4*)&As[r * LDS_STRIDE + c4] = v;
    }
    __syncthreads();

    const int lane = tid & 31;
    const int wave = tid >> 5;
    const int n0   = wave * 64;
    const int m    = lane & 15;
    const int koff = (lane >> 4) << 1;  // 0 for lanes 0-15, 2 for lanes 16-31

    v8f acc0 = {}, acc1 = {}, acc2 = {}, acc3 = {};

    for (int kk = 0; kk < IN_CH; kk += 4) {
        // A fragment: two consecutive K values for this lane's (m, koff).
        v2f a = *(const v2f*)&As[m * LDS_STRIDE + kk + koff];

        // B fragments: rows (kk+koff, kk+koff+1) of W, 16 columns per tile.
        const float* bp = W + (size_t)(kk + koff) * HID_CH + n0 + (lane & 15);
        v2f b0; b0.x = bp[0];      b0.y = bp[HID_CH];
        v2f b1; b1.x = bp[16];     b1.y = bp[HID_CH + 16];
        v2f b2; b2.x = bp[32];     b2.y = bp[HID_CH + 32];
        v2f b3; b3.x = bp[48];     b3.y = bp[HID_CH + 48];

        acc0 = __builtin_amdgcn_wmma_f32_16x16x4_f32(false, a, false, b0,
                                                     (short)0, acc0, false, false);
        acc1 = __builtin_amdgcn_wmma_f32_16x16x4_f32(false, a, false, b1,
                                                     (short)0, acc1, false, false);
        acc2 = __builtin_amdgcn_wmma_f32_16x16x4_f32(false, a, false, b2,
                                                     (short)0, acc2, false, false);
        acc3 = __builtin_amdgcn_wmma_f32_16x16x4_f32(false, a, false, b3,
                                                     (short)0, acc3, false, false);
    }

    // Epilogue: bias + PReLU, store.
    const float alpha = alpha_p[0];
    const int rbase = (lane < 16) ? 0 : 8;
    v8f accs[4] = {acc0, acc1, acc2, acc3};
#pragma unroll
    for (int t = 0; t < 4; ++t) {
        const int col = n0 + t * 16 + (lane & 15);
        const float bb = bias[col];
#pragma unroll
        for (int r = 0; r < 8; ++r) {
            float h = accs[t][r] + bb;
            float y = fmaxf(h, 0.0f) + alpha * fminf(h, 0.0f);
            out[(size_t)(row0 + rbase + r) * HID_CH + col] = y;
        }
    }
}

// ---------------------------------------------------------------------------
// Launch
// ---------------------------------------------------------------------------
extern "C" void kernel_launch(void* const* d_in, const int* in_sizes, int n_in,
                              void* d_out, int out_size, void* d_ws, size_t ws_size,
                              hipStream_t stream) {
    const float*     x     = (const float*)d_in[0];
    const long long* ei    = (const long long*)d_in[1];  // int64 per reference
    const float*     W     = (const float*)d_in[2];
    const float*     bias  = (const float*)d_in[3];
    const float*     alpha = (const float*)d_in[4];
    float*           out   = (float*)d_out;

    const int nNodes = in_sizes[0] / IN_CH;   // 100000
    const int nE     = in_sizes[1] / 2;       // 1600000
    const long long* src = ei;
    const long long* dst = ei + nE;

    // Workspace: agg [nNodes x 128] f32, then deg/dinv [nNodes] f32.
    float* agg = (float*)d_ws;
    float* deg = agg + (size_t)nNodes * IN_CH;

    deg_init_kernel<<<(nNodes + 255) / 256, 256, 0, stream>>>(deg, nNodes);
    deg_accum_kernel<<<(nE + 255) / 256, 256, 0, stream>>>(dst, deg, nE);
    deg_rsqrt_kernel<<<(nNodes + 255) / 256, 256, 0, stream>>>(deg, nNodes);

    {
        size_t threads = (size_t)nNodes * 32;
        self_init_kernel<<<(unsigned)((threads + 255) / 256), 256, 0, stream>>>(
            x, deg, agg, nNodes);
    }
    {
        size_t threads = (size_t)nE * 32;
        edge_scatter_kernel<<<(unsigned)((threads + 255) / 256), 256, 0, stream>>>(
            x, src, dst, deg, agg, nE);
    }

    // nNodes is an exact multiple of 16 (100000 = 6250 * 16).
    gemm_prelu_kernel<<<nNodes / 16, 256, 0, stream>>>(agg, W, bias, alpha, out,
                                                       nNodes);
}